// PermutohedralLattice_9070970929355
// MI455X (gfx1250) — compile-verified
//
#include <hip/hip_runtime.h>

typedef __attribute__((ext_vector_type(8))) float v8f;
typedef __attribute__((ext_vector_type(2))) float v2f;

// ---------------------------------------------------------------- zero
__global__ void k_zero(float* __restrict__ p, long long n) {
  long long i = (long long)blockIdx.x * blockDim.x + threadIdx.x;
  long long stride = (long long)gridDim.x * blockDim.x;
  for (; i < n; i += stride) p[i] = 0.0f;
}

// ---------------------------------------------------------------- pass A: splat ones (C=1)
// acc[simp[n,e]+1] += bw[n,e]
__global__ void k_splat1(const float* __restrict__ bw, const int* __restrict__ simp,
                         float* __restrict__ lat, int total) {
  int i = blockIdx.x * blockDim.x + threadIdx.x;
  if (i >= total) return;
  unsafeAtomicAdd(&lat[simp[i] + 1], bw[i]);
}

// ---------------------------------------------------------------- blur (C=1), Jacobi
__global__ void k_blur1(const float* __restrict__ src, float* __restrict__ dst,
                        const int* __restrict__ nbr, int M, int ED, int d) {
  int m = blockIdx.x * blockDim.x + threadIdx.x;
  if (m >= M) return;
  int n0 = nbr[(m * 2 + 0) * ED + d];   // -1 -> row 0 (zero row)
  int n1 = nbr[(m * 2 + 1) * ED + d];
  dst[m + 1] = src[m + 1] + 0.5f * (src[n0 + 1] + src[n1 + 1]);
}

// ---------------------------------------------------------------- slice (C=1) -> norm
__global__ void k_slice_norm(const float* __restrict__ bw, const int* __restrict__ simp,
                             const float* __restrict__ lat, float* __restrict__ norm,
                             int N, int ED, float alpha) {
  int n = blockIdx.x * blockDim.x + threadIdx.x;
  if (n >= N) return;
  float s = 0.0f;
  for (int e = 0; e < ED; ++e)
    s += bw[n * ED + e] * lat[simp[n * ED + e] + 1];
  norm[n] = rsqrtf(alpha * s + 1e-20f);
}

// ---------------------------------------------------------------- pass B: splat y*norm (C=16)
__global__ void k_splat16(const float* __restrict__ y, const float* __restrict__ bw,
                          const int* __restrict__ simp, const float* __restrict__ norm,
                          float* __restrict__ lat, int N, int ED) {
  int t = blockIdx.x * blockDim.x + threadIdx.x;
  if (t >= N * 16) return;
  int n = t >> 4, c = t & 15;            // 16 consecutive lanes share n -> broadcast bw/simp
  float v = y[t] * norm[n];
  for (int e = 0; e < ED; ++e) {
    int s = simp[n * ED + e] + 1;
    unsafeAtomicAdd(&lat[(long long)s * 16 + c], bw[n * ED + e] * v);
  }
}

// ---------------------------------------------------------------- blur (C=16), Jacobi
__global__ void k_blur16(const float* __restrict__ src, float* __restrict__ dst,
                         const int* __restrict__ nbr, int M, int ED, int d) {
  long long t = (long long)blockIdx.x * blockDim.x + threadIdx.x;
  if (t >= (long long)M * 16) return;
  int m = (int)(t >> 4), c = (int)(t & 15);
  int n0 = nbr[(m * 2 + 0) * ED + d];
  int n1 = nbr[(m * 2 + 1) * ED + d];
  dst[(long long)(m + 1) * 16 + c] =
      src[(long long)(m + 1) * 16 + c] +
      0.5f * (src[(long long)(n0 + 1) * 16 + c] + src[(long long)(n1 + 1) * 16 + c]);
}

// ---------------------------------------------------------------- slice (C=16) via WMMA
// One wave per tile of 16 points. out_tile(16x16) = sum_e diag(bw[:,e]) @ G_e,
// where G_e[r,c] = lat[(simp[n0+r,e]+1)*16 + c]. diag(w)@G is decomposed exactly
// into four V_WMMA_F32_16X16X4_F32 (K-chunks of the diagonal) — full f32 precision.
//
// Layouts (ISA 7.12.2, wave32):
//   A 16x4 f32 : 2 VGPRs, lane L -> A[L%16][K = vgpr + 2*(L>=16)]
//   B 4x16 f32 : 2 VGPRs, lane L -> B[K = vgpr + 2*(L>=16)][N = L%16]
//   D 16x16    : acc[v]  -> D[M = v + 8*(L>=16)][N = L%16]
__global__ void __launch_bounds__(256)
k_slice16_wmma(const float* __restrict__ bw, const int* __restrict__ simp,
               const float* __restrict__ lat, const float* __restrict__ norm,
               float* __restrict__ out, int ntiles, int ED, float alpha) {
  int wave = blockIdx.x * (blockDim.x >> 5) + (threadIdx.x >> 5);
  if (wave >= ntiles) return;            // wave-uniform: EXEC all ones below
  int lane = threadIdx.x & 31;
  int m  = lane & 15;                    // column index / point-in-tile for A
  int hi = lane >> 4;
  int n0 = wave * 16;

  v8f acc = {};
  for (int e = 0; e < ED; ++e) {
    float w = bw[(n0 + m) * ED + e];     // diagonal entry for row m of the tile
#pragma unroll
    for (int j = 0; j < 4; ++j) {
      // A chunk: columns 4j..4j+3 of diag(w); lane holds K = 2*hi, 2*hi+1
      v2f a;
      a.x = (m == 4 * j + 2 * hi)     ? w : 0.0f;
      a.y = (m == 4 * j + 2 * hi + 1) ? w : 0.0f;
      // B chunk: gathered lattice rows r0 = 4j+2hi, r0+1 ; col = m (coalesced 64B/half-wave)
      int r0 = 4 * j + 2 * hi;
      int s0 = simp[(n0 + r0) * ED + e] + 1;
      int s1 = simp[(n0 + r0 + 1) * ED + e] + 1;
      v2f b;
      b.x = lat[(long long)s0 * 16 + m];
      b.y = lat[(long long)s1 * 16 + m];
      acc = __builtin_amdgcn_wmma_f32_16x16x4_f32(false, a, false, b,
                                                  (short)0, acc, false, false);
    }
  }
#pragma unroll
  for (int v = 0; v < 8; ++v) {
    int row = n0 + v + 8 * hi;
    out[(long long)row * 16 + m] = acc[v] * alpha * norm[row];
  }
}

// scalar tail (N % 16 != 0)
__global__ void k_slice16_scalar(const float* __restrict__ bw, const int* __restrict__ simp,
                                 const float* __restrict__ lat, const float* __restrict__ norm,
                                 float* __restrict__ out, int nstart, int N, int ED, float alpha) {
  int t = nstart * 16 + blockIdx.x * blockDim.x + threadIdx.x;
  if (t >= N * 16) return;
  int n = t >> 4, c = t & 15;
  float s = 0.0f;
  for (int e = 0; e < ED; ++e)
    s += bw[n * ED + e] * lat[(long long)(simp[n * ED + e] + 1) * 16 + c];
  out[t] = s * alpha * norm[n];
}

// ================================================================ host
extern "C" void kernel_launch(void* const* d_in, const int* in_sizes, int n_in,
                              void* d_out, int out_size, void* d_ws, size_t ws_size,
                              hipStream_t stream) {
  const float* y    = (const float*)d_in[0];
  const float* bw   = (const float*)d_in[1];
  const int*   simp = (const int*)d_in[2];
  const int*   nbr  = (const int*)d_in[3];
  float*       outp = (float*)d_out;

  const int C  = 16;
  const int N  = in_sizes[0] / C;
  const int ED = in_sizes[1] / N;                 // 6
  const int M  = in_sizes[3] / (2 * ED);
  const float alpha = 1.0f / (1.0f + ldexpf(1.0f, -(ED - 1)));

  // workspace carve-out (256B aligned)
  size_t off = 0;
  auto carve = [&](size_t nfloats) {
    float* p = (float*)((char*)d_ws + off);
    off += ((nfloats * sizeof(float) + 255) / 256) * 256;
    return p;
  };
  float* latA  = carve((size_t)M + 1);
  float* latB  = carve((size_t)M + 1);
  float* normv = carve((size_t)N);
  float* lat0  = carve(((size_t)M + 1) * 16);
  float* lat1  = carve(((size_t)M + 1) * 16);
  (void)ws_size;

  const int T = 256;
  // zero the whole carved region (includes the mandatory zero row 0 of all buffers)
  long long zn = (long long)(off / sizeof(float));
  k_zero<<<2048, T, 0, stream>>>((float*)d_ws, zn);

  // ---- pass A: filter(ones) -> norm
  int tot1 = N * ED;
  k_splat1<<<(tot1 + T - 1) / T, T, 0, stream>>>(bw, simp, latA, tot1);
  {
    float* s = latA; float* d = latB;
    for (int dd = 0; dd < ED; ++dd) {
      k_blur1<<<(M + T - 1) / T, T, 0, stream>>>(s, d, nbr, M, ED, dd);
      float* tmp = s; s = d; d = tmp;
    }
    k_slice_norm<<<(N + T - 1) / T, T, 0, stream>>>(bw, simp, s, normv, N, ED, alpha);
  }

  // ---- pass B: filter(y * norm) -> out, scaled by alpha*norm
  int tot16 = N * 16;
  k_splat16<<<(tot16 + T - 1) / T, T, 0, stream>>>(y, bw, simp, normv, lat0, N, ED);
  {
    float* s = lat0; float* d = lat1;
    long long mt = (long long)M * 16;
    int blocks = (int)((mt + T - 1) / T);
    for (int dd = 0; dd < ED; ++dd) {
      k_blur16<<<blocks, T, 0, stream>>>(s, d, nbr, M, ED, dd);
      float* tmp = s; s = d; d = tmp;
    }
    int ntiles = N / 16;
    if (ntiles > 0) {
      int wavesPerBlock = T / 32;  // 8 tiles per block
      int gb = (ntiles + wavesPerBlock - 1) / wavesPerBlock;
      k_slice16_wmma<<<gb, T, 0, stream>>>(bw, simp, s, normv, outp, ntiles, ED, alpha);
    }
    int rem = N - (N / 16) * 16;
    if (rem > 0) {
      int rt = rem * 16;
      k_slice16_scalar<<<(rt + T - 1) / T, T, 0, stream>>>(bw, simp, s, normv, outp,
                                                           (N / 16) * 16, N, ED, alpha);
    }
  }
}